// DifferentiableSimulator_61641370632767
// MI455X (gfx1250) — compile-verified
//
#include <hip/hip_runtime.h>
#include <math.h>

typedef __attribute__((ext_vector_type(2))) float v2f;
typedef __attribute__((ext_vector_type(8))) float v8f;

#define DEG2RAD 0.017453292519943295f
#define LOG2E   1.4426950408889634f

// native v_exp_f32 (base-2 exponential)
__device__ __forceinline__ float fast_exp2(float x) {
    return __builtin_amdgcn_exp2f(x);
}

// ---------------------------------------------------------------------------
// Kernel 0: template mean (3 floats) + zero per-batch max slots
// ---------------------------------------------------------------------------
__global__ void k_init(const float* __restrict__ tmpl, int N,
                       float* __restrict__ tmean, float* __restrict__ maxbuf, int B)
{
    __shared__ float sx[256], sy[256], sz[256];
    int t = threadIdx.x;
    float ax = 0.f, ay = 0.f, az = 0.f;
    for (int n = t; n < N; n += 256) {
        ax += tmpl[n*3+0]; ay += tmpl[n*3+1]; az += tmpl[n*3+2];
    }
    sx[t] = ax; sy[t] = ay; sz[t] = az;
    __syncthreads();
    for (int s = 128; s > 0; s >>= 1) {
        if (t < s) { sx[t] += sx[t+s]; sy[t] += sy[t+s]; sz[t] += sz[t+s]; }
        __syncthreads();
    }
    if (t == 0) {
        float inv = 1.0f / (float)N;
        tmean[0] = sx[0]*inv; tmean[1] = sy[0]*inv; tmean[2] = sz[0]*inv;
    }
    if (t < B) maxbuf[t] = 0.0f;
}

// ---------------------------------------------------------------------------
// Kernel 1: per-contact geometry -> contact_pos (x,y,z), |p|^2, sigmoid(logits)
// Per-batch scalar work (rotation, LUT bilerp) is recomputed per thread (cheap).
// ---------------------------------------------------------------------------
__global__ void k_prep(const float* __restrict__ params,
                       const float* __restrict__ logits,
                       const float* __restrict__ tmpl,
                       const float* __restrict__ start_loc,
                       const float* __restrict__ lut,
                       const float* __restrict__ agrid,
                       const float* __restrict__ bgrid,
                       const float* __restrict__ tmean,
                       int B, int N, int NA, int NB,
                       float* __restrict__ px_, float* __restrict__ py_,
                       float* __restrict__ pz_, float* __restrict__ p2_,
                       float* __restrict__ ep_)
{
    int idx = blockIdx.x * blockDim.x + threadIdx.x;
    if (idx >= B*N) return;
    int b = idx / N, n = idx - b*N;

    float alpha  = params[b*4+0];
    float beta   = params[b*4+1];
    float offset = params[b*4+2];
    float shank  = params[b*4+3];

    float a  = alpha * DEG2RAD, bb = beta * DEG2RAD;
    float sa = __sinf(a),  ca = __cosf(a);
    float sb = __sinf(bb), cb = __cosf(bb);
    // R = Rx(alpha) @ Ry(beta)
    float R00 =  cb,     R01 = 0.f, R02 =  sb;
    float R10 =  sa*sb,  R11 = ca,  R12 = -sa*cb;
    float R20 = -ca*sb,  R21 = sa,  R22 =  ca*cb;
    // direction = R @ [0,0,-1] = -R[:,2]  (unit by construction; normalize anyway)
    float dx = -R02, dy = -R12, dz = -R22;
    float dn = rsqrtf(dx*dx + dy*dy + dz*dz);
    dx *= dn; dy *= dn; dz *= dn;

    // bilinear surface-distance LUT interp
    float ag0 = agrid[0], agL = agrid[NA-1];
    float bg0 = bgrid[0], bgL = bgrid[NB-1];
    float ai = (alpha - ag0) / (agL - ag0 + 1e-8f) * (float)(NA-1);
    float bi = (beta  - bg0) / (bgL - bg0 + 1e-8f) * (float)(NB-1);
    ai = fminf(fmaxf(ai, 0.f), (float)(NA-1));
    bi = fminf(fmaxf(bi, 0.f), (float)(NB-1));
    int a0 = min(max((int)floorf(ai), 0), NA-1);
    int b0 = min(max((int)floorf(bi), 0), NB-1);
    int a1 = min(a0+1, NA-1), b1 = min(b0+1, NB-1);
    float fa = ai - (float)a0, fb = bi - (float)b0;
    float v00 = lut[a0*NB+b0], v01 = lut[a0*NB+b1];
    float v10 = lut[a1*NB+b0], v11 = lut[a1*NB+b1];
    float sd = v00*(1.f-fa)*(1.f-fb) + v01*(1.f-fa)*fb + v10*fa*(1.f-fb) + v11*fa*fb;
    sd = fmaxf(sd, 1.0f);

    float pen = sd - shank*0.5f - offset;
    float gcx = start_loc[0] + dx*pen;
    float gcy = start_loc[1] + dy*pen;
    float gcz = start_loc[2] + dz*pen;

    float gx0 = tmpl[n*3+0] - tmean[0];
    float gy0 = tmpl[n*3+1] - tmean[1];
    float gz0 = (tmpl[n*3+2] - tmean[2]) * shank;

    float rx = R00*gx0 + R01*gy0 + R02*gz0;
    float ry = R10*gx0 + R11*gy0 + R12*gz0;
    float rz = R20*gx0 + R21*gy0 + R22*gz0;

    float cx = rx + gcx, cy = ry + gcy, cz = rz + gcz;
    px_[idx] = cx; py_[idx] = cy; pz_[idx] = cz;
    p2_[idx] = cx*cx + cy*cy + cz*cz;
    ep_[idx] = 1.0f / (1.0f + __expf(-logits[idx]));
}

// ---------------------------------------------------------------------------
// Kernel 2: soft match via V_WMMA_F32_16X16X4_F32.
//   d2[m, v] = p2[m] + |v|^2 - 2 p.v  ==  A(16x4) * B(4x16) + C  where
//   A row m = [x, y, z, 1],  B col v = [-2vx, -2vy, -2vz, |v|^2],  C = p2[m].
// Each wave owns 16 contact rows and streams all V in 16-column WMMA tiles,
// accumulating wsum / w*pol / w*ecc per D register, butterfly-reduced at end.
// ---------------------------------------------------------------------------
__global__ void k_softmatch(const float* __restrict__ px_, const float* __restrict__ py_,
                            const float* __restrict__ pz_, const float* __restrict__ p2_,
                            const float* __restrict__ v1_pos, const float* __restrict__ v1_prf,
                            int M, int V, int ntiles,
                            float* __restrict__ wsum, float* __restrict__ polsum,
                            float* __restrict__ eccsum)
{
    int wave = threadIdx.x >> 5;
    int lane = threadIdx.x & 31;
    int tile = blockIdx.x * (blockDim.x >> 5) + wave;
    if (tile >= ntiles) return;

    int half = lane >> 4;     // 0: K=0/1 , 1: K=2/3
    int l16  = lane & 15;
    int rowA = tile*16 + l16;
    bool rv  = rowA < M;

    float ax = rv ? px_[rowA] : 0.f;
    float ay = rv ? py_[rowA] : 0.f;
    float az = rv ? pz_[rowA] : 0.f;

    // A-matrix 16x4 f32: lanes 0-15 hold K=0 (VGPR0) / K=1 (VGPR1),
    //                    lanes 16-31 hold K=2 / K=3 for the same rows.
    v2f A;
    A.x = half ? az : ax;
    A.y = half ? 1.0f : ay;

    // C 16x16 f32: VGPR k, lanes 0-15 -> M=k ; lanes 16-31 -> M=k+8 (broadcast over N)
    v8f C;
#pragma unroll
    for (int k = 0; k < 8; ++k) {
        int r = tile*16 + k + 8*half;
        C[k] = (r < M) ? p2_[r] : 1e30f;
    }

    const float negfac = -LOG2E / (2.0f * 1.5f * 1.5f);   // exp(-d2/4.5) via exp2
    float sw[8], sp[8], se[8];
#pragma unroll
    for (int k = 0; k < 8; ++k) { sw[k] = 0.f; sp[k] = 0.f; se[k] = 0.f; }

    for (int v0 = 0; v0 < V; v0 += 16) {
        if (v0 + 256 < V) {
            __builtin_prefetch(v1_pos + 3*(v0 + 256), 0, 0);  // global_prefetch_b8
            __builtin_prefetch(v1_prf + 3*(v0 + 256), 0, 0);
        }
        int v = v0 + l16;
        float vx, vy, vz, pol, ec;
        if (v < V) {
            vx = v1_pos[v*3+0]; vy = v1_pos[v*3+1]; vz = v1_pos[v*3+2];
            pol = v1_prf[v*3+0]; ec = v1_prf[v*3+1];
        } else {                       // dead column: push d2 huge -> w = 0
            vx = vy = vz = 1e6f; pol = 0.f; ec = 0.f;
        }
        // B-matrix 4x16 f32: lanes 0-15 hold K=0/K=1, lanes 16-31 hold K=2/K=3, N=l16
        v2f Bm;
        Bm.x = half ? (-2.0f*vz) : (-2.0f*vx);
        Bm.y = half ? (vx*vx + vy*vy + vz*vz) : (-2.0f*vy);

        v8f D = __builtin_amdgcn_wmma_f32_16x16x4_f32(
            /*neg_a=*/false, A, /*neg_b=*/false, Bm,
            /*c_mod=*/(short)0, C, /*reuse_a=*/false, /*reuse_b=*/false);

#pragma unroll
        for (int k = 0; k < 8; ++k) {
            float d2 = fmaxf(D[k], 0.0f);
            float w  = fast_exp2(d2 * negfac);
            sw[k] += w;
            sp[k] += w * pol;       // lane's column v supplies pol/ecc
            se[k] += w * ec;
        }
    }

    // reduce over the 16 N-lanes of each half (xor masks < 16 stay in-half)
#pragma unroll
    for (int k = 0; k < 8; ++k) {
        for (int m = 1; m < 16; m <<= 1) {
            sw[k] += __shfl_xor(sw[k], m, 32);
            sp[k] += __shfl_xor(sp[k], m, 32);
            se[k] += __shfl_xor(se[k], m, 32);
        }
    }
    if (l16 == 0) {
#pragma unroll
        for (int k = 0; k < 8; ++k) {
            int r = tile*16 + k + 8*half;
            if (r < M) { wsum[r] = sw[k]; polsum[r] = sp[k]; eccsum[r] = se[k]; }
        }
    }
}

// ---------------------------------------------------------------------------
// Kernel 3: per-contact phosphene parameters -> float4 (cx, cy, -log2e/s^2, w)
// ---------------------------------------------------------------------------
__global__ void k_contact(const float* __restrict__ wsum, const float* __restrict__ polsum,
                          const float* __restrict__ eccsum, const float* __restrict__ ep,
                          int M, float4* __restrict__ cparams)
{
    int i = blockIdx.x * blockDim.x + threadIdx.x;
    if (i >= M) return;
    float ws = wsum[i];
    float validity = fminf(ws, 1.0f);
    float inv = 1.0f / (ws + 1e-8f);
    float pol = polsum[i] * inv;
    float ecc = eccsum[i] * inv;
    float ang = pol * DEG2RAD;
    float m   = 17.3f * (1.0f/(ecc + 0.75f) - 1.0f/(ecc + 120.0f));
    float m_inv = 1.0f / (fabsf(m) + 1e-8f);
    const float SPREAD = 0.38490018f;            // sqrt(100/675)
    float psig = SPREAD * m_inv * 0.5f;
    const float scaled = 256.0f / 90.0f;         // MAP_SIZE / VIEW_ANGLE
    float sn, cs;
    __sincosf(ang, &sn, &cs);
    float cx = ecc * cs * scaled + 128.0f;
    float cy = ecc * sn * scaled + 128.0f;
    float ps = fmaxf(psig * scaled, 1.0f);
    float nf = -LOG2E / (ps*ps + 1e-8f);
    float w  = ep[i] * validity;
    cparams[i] = make_float4(cx, cy, nf, w);
}

// ---------------------------------------------------------------------------
// Kernel 4: render with rot90 baked in. out[b,i,j] = sum_c w*exp2(nf*r2) with
// gx = 255-i, gy = j. Contacts staged through LDS (broadcast to whole block).
// Block max -> int atomicMax (all values >= 0).
// ---------------------------------------------------------------------------
__global__ void k_render(const float4* __restrict__ cparams, int N, int H,
                         float* __restrict__ outmap, float* __restrict__ maxbuf)
{
    int j = threadIdx.x;           // gy
    int i = blockIdx.x;            // output row
    int b = blockIdx.y;
    float px = (float)(H - 1 - i); // gx after rot90(k=1)
    float py = (float)j;

    __shared__ float4 sc[128];
    __shared__ float  smax[256];

    const float4* base = cparams + (size_t)b * N;
    float acc = 0.f;
    for (int c0 = 0; c0 < N; c0 += 128) {
        int nc = min(128, N - c0);
        if (threadIdx.x < nc) sc[threadIdx.x] = base[c0 + threadIdx.x];
        __syncthreads();
        for (int c = 0; c < nc; ++c) {
            float4 p = sc[c];
            float dx = px - p.x;
            float dy = py - p.y;
            float r2 = fmaf(dx, dx, dy*dy);
            acc += p.w * fast_exp2(p.z * r2);
        }
        __syncthreads();
    }
    outmap[((size_t)b * H + i) * H + j] = acc;

    smax[threadIdx.x] = acc;
    __syncthreads();
    for (int s = 128; s > 0; s >>= 1) {
        if (threadIdx.x < s) smax[threadIdx.x] = fmaxf(smax[threadIdx.x], smax[threadIdx.x+s]);
        __syncthreads();
    }
    if (threadIdx.x == 0)
        atomicMax((int*)&maxbuf[b], __float_as_int(smax[0]));
}

// ---------------------------------------------------------------------------
// Kernel 5: in-place normalize by per-batch max
// ---------------------------------------------------------------------------
__global__ void k_norm(float* __restrict__ outmap, const float* __restrict__ maxbuf,
                       int total, int perB)
{
    int idx = blockIdx.x * blockDim.x + threadIdx.x;
    if (idx >= total) return;
    int b = idx / perB;
    outmap[idx] = outmap[idx] / (maxbuf[b] + 1e-8f);
}

// ---------------------------------------------------------------------------
extern "C" void kernel_launch(void* const* d_in, const int* in_sizes, int n_in,
                              void* d_out, int out_size, void* d_ws, size_t ws_size,
                              hipStream_t stream)
{
    const float* params    = (const float*)d_in[0];
    const float* logits    = (const float*)d_in[1];
    const float* v1_pos    = (const float*)d_in[2];
    const float* v1_prf    = (const float*)d_in[3];
    const float* start_loc = (const float*)d_in[4];
    const float* lut       = (const float*)d_in[5];
    const float* agrid     = (const float*)d_in[6];
    const float* bgrid     = (const float*)d_in[7];
    const float* tmpl      = (const float*)d_in[8];

    int B  = in_sizes[0] / 4;
    int V  = in_sizes[2] / 3;
    int N  = in_sizes[8] / 3;
    int NA = in_sizes[6];
    int NB = in_sizes[7];
    int M  = B * N;
    const int H = 256;

    float* ws = (float*)d_ws;
    float* px_    = ws + 0*M;
    float* py_    = ws + 1*M;
    float* pz_    = ws + 2*M;
    float* p2_    = ws + 3*M;
    float* ep_    = ws + 4*M;
    float* wsum   = ws + 5*M;
    float* polsum = ws + 6*M;
    float* eccsum = ws + 7*M;
    float4* cparams = (float4*)(ws + 8*M);   // 4*M floats
    float* maxbuf = ws + 12*M;
    float* tmean  = ws + 12*M + B;
    float* outmap = (float*)d_out;

    k_init<<<1, 256, 0, stream>>>(tmpl, N, tmean, maxbuf, B);

    k_prep<<<(M + 255)/256, 256, 0, stream>>>(params, logits, tmpl, start_loc, lut,
                                              agrid, bgrid, tmean, B, N, NA, NB,
                                              px_, py_, pz_, p2_, ep_);

    int ntiles = (M + 15) / 16;
    k_softmatch<<<(ntiles + 7)/8, 256, 0, stream>>>(px_, py_, pz_, p2_, v1_pos, v1_prf,
                                                    M, V, ntiles, wsum, polsum, eccsum);

    k_contact<<<(M + 255)/256, 256, 0, stream>>>(wsum, polsum, eccsum, ep_, M, cparams);

    dim3 rgrid(H, B);
    k_render<<<rgrid, 256, 0, stream>>>(cparams, N, H, outmap, maxbuf);

    int total = B * H * H;
    k_norm<<<(total + 255)/256, 256, 0, stream>>>(outmap, maxbuf, total, H*H);
}